// SimLoss_46471546143020
// MI455X (gfx1250) — compile-verified
//
#include <hip/hip_runtime.h>
#include <math.h>

// ---------------------------------------------------------------------------
// SimLoss (NT-Xent style) fused for MI455X / gfx1250.
//   x = reorder(batch)  [8192,256] f32
//   den[k] = sum_j exp(sim[k,j]/(n_k n_j + eps))   (diag term == 1)
//   num[k] = sim[i,i+1]/(n_i n_j), i = 2*(k/2)
//   loss   = sum_k (log den[k] - num[k]/tau) / n
//
// sim = x x^T is symmetric: only upper-triangular 128x128 blocks (bx>=by) are
// computed with V_WMMA_F32_16X16X4_F32; each off-diagonal tile's exp() values
// feed both row sums (slot bx) and column sums (slot by) of a single-writer
// partial matrix partial[8192][64] -> atomic-free, bitwise-deterministic.
// ---------------------------------------------------------------------------

typedef __attribute__((ext_vector_type(2))) float v2f;
typedef __attribute__((ext_vector_type(8))) float v8f;

#define N_ROWS 8192
#define DIM    256
#define HALF   4096
#define NBLK   64          // 8192 / 128 block-rows/cols
#define TAU_F  1.0f
#define EPS_F  1e-6f

// --- x[row] = batch[map(row)], nrm[row] = ||x[row]||  (one block per row) ---
__global__ __launch_bounds__(256) void sim_reorder_norm_kernel(
    const float* __restrict__ batch, float* __restrict__ x, float* __restrict__ nrm) {
    const int row = blockIdx.x;
    const int tid = threadIdx.x;                 // 256 threads == DIM
    const int src = (row & 1) ? (HALF + (row >> 1)) : (row >> 1);
    const float v = batch[(size_t)src * DIM + tid];
    x[(size_t)row * DIM + tid] = v;

    float sq = v * v;
    #pragma unroll
    for (int off = 16; off > 0; off >>= 1) sq += __shfl_xor(sq, off, 32);

    __shared__ float red[8];
    const int wave = tid >> 5, lane = tid & 31;
    if (lane == 0) red[wave] = sq;
    __syncthreads();
    if (tid == 0) {
        float t = 0.0f;
        #pragma unroll
        for (int i = 0; i < 8; ++i) t += red[i];
        nrm[row] = sqrtf(t);
    }
}

// --- fused GEMM (f32 WMMA) + exp, upper-triangular blocks only --------------
// Block = 256 threads (8 waves). Block tile: 128x128 of sim, bx >= by.
// Wave w: rows m0 = by*128 + w*16, cols n0 = bx*128 .. +127 (8 16x16 tiles).
// Row sums   -> partial[row*64 + bx]   (single writer)
// Col sums   -> partial[col*64 + by]   (single writer, off-diag blocks only)
__global__ __launch_bounds__(256) void sim_partial_kernel(
    const float* __restrict__ x, const float* __restrict__ nrm,
    float* __restrict__ partial) {
    if (blockIdx.x < blockIdx.y) return;         // lower triangle: nothing to do
    const bool offdiag = (blockIdx.x != blockIdx.y);

    const int lane = threadIdx.x & 31;
    const int wave = threadIdx.x >> 5;
    const int m0 = blockIdx.y * 128 + wave * 16;
    const int n0 = blockIdx.x * 128;
    const int l15  = lane & 15;
    const int koff = (lane >> 4) * 2;   // lanes 16-31 hold K=+2,+3 (ISA 16x4 layout)

    // f32 A 16x4: lane l -> row m0+(l&15), cols {koff, koff+1}  == one float2
    // f32 B 4x16: identical form with rows n0+t*16+(l&15) (B = x^T)
    const float* Ap = x + (size_t)(m0 + l15) * DIM + koff;
    const float* Bp = x + (size_t)(n0 + l15) * DIM + koff;

    v8f acc[8];
    #pragma unroll
    for (int t = 0; t < 8; ++t) acc[t] = (v8f){0, 0, 0, 0, 0, 0, 0, 0};

    for (int s = 0; s < DIM / 4; ++s) {          // 64 K-steps of 4
        const v2f a = *(const v2f*)(Ap + 4 * s);
        #pragma unroll
        for (int t = 0; t < 8; ++t) {
            const v2f b = *(const v2f*)(Bp + (size_t)t * 16 * DIM + 4 * s);
            acc[t] = __builtin_amdgcn_wmma_f32_16x16x4_f32(
                false, a, false, b, (short)0, acc[t], false, false);
        }
    }

    // C/D layout: acc[t][r] at lane l = sim[m0 + r + (l>=16?8:0)][n0 + t*16 + (l&15)]
    const int hi8 = (lane >> 4) * 8;
    float nr[8];
    #pragma unroll
    for (int r = 0; r < 8; ++r) nr[r] = nrm[m0 + hi8 + r];

    float rowsum[8] = {0, 0, 0, 0, 0, 0, 0, 0};  // per r (VGPR row)
    float colsum[8] = {0, 0, 0, 0, 0, 0, 0, 0};  // per t (column tile)
    #pragma unroll
    for (int t = 0; t < 8; ++t) {
        const int col = n0 + t * 16 + l15;
        const float nc = nrm[col];
        #pragma unroll
        for (int r = 0; r < 8; ++r) {
            const int row = m0 + hi8 + r;
            float e;
            if (offdiag) {                        // row != col guaranteed
                e = __expf(acc[t][r] / (nr[r] * nc + EPS_F) * (1.0f / TAU_F));
            } else {
                e = (row == col)
                    ? 1.0f                        // masked diag -> exp(0)
                    : __expf(acc[t][r] / (nr[r] * nc + EPS_F) * (1.0f / TAU_F));
            }
            rowsum[r] += e;
            colsum[t] += e;
        }
    }

    // --- row sums: reduce across the 16 lanes sharing each row, store slot bx
    #pragma unroll
    for (int r = 0; r < 8; ++r) {
        float s = rowsum[r];
        s += __shfl_xor(s, 1, 32);
        s += __shfl_xor(s, 2, 32);
        s += __shfl_xor(s, 4, 32);
        s += __shfl_xor(s, 8, 32);
        if (l15 == 0)
            partial[(size_t)(m0 + hi8 + r) * NBLK + blockIdx.x] = s;
    }

    // --- col sums (off-diagonal only): combine halves, then waves via LDS ---
    if (offdiag) {
        __shared__ float colw[8][128];
        #pragma unroll
        for (int t = 0; t < 8; ++t) {
            float cs = colsum[t] + __shfl_xor(colsum[t], 16, 32); // both 8-row halves
            if (lane < 16) colw[wave][t * 16 + l15] = cs;
        }
        __syncthreads();
        if (threadIdx.x < 128) {
            float s = 0.0f;
            #pragma unroll
            for (int w = 0; w < 8; ++w) s += colw[w][threadIdx.x]; // fixed order
            partial[(size_t)(n0 + threadIdx.x) * NBLK + blockIdx.y] = s;
        }
    }
}

// --- den[k] = fixed-order sum of the 64 single-writer partials --------------
__global__ __launch_bounds__(256) void sim_den_reduce_kernel(
    const float* __restrict__ partial, float* __restrict__ den) {
    const int k = blockIdx.x * blockDim.x + threadIdx.x;
    const float* p = partial + (size_t)k * NBLK;
    float s = 0.0f;
    #pragma unroll
    for (int b = 0; b < NBLK; ++b) s += p[b];
    den[k] = s;
}

// --- per-pair term: log den[2p] + log den[2p+1] - 2*num/tau -----------------
// One wave per pair (i, i+1); 8 waves per block. No atomics.
__global__ __launch_bounds__(256) void sim_pair_kernel(
    const float* __restrict__ x, const float* __restrict__ nrm,
    const float* __restrict__ den, float* __restrict__ pairterm) {
    const int lane = threadIdx.x & 31;
    const int wave = threadIdx.x >> 5;
    const int pair = blockIdx.x * 8 + wave;      // 0..4095
    const int i = pair * 2;

    const float* xi = x + (size_t)i * DIM;
    const float* xj = xi + DIM;
    float d = 0.0f;
    #pragma unroll
    for (int c = lane; c < DIM; c += 32) d += xi[c] * xj[c];
    #pragma unroll
    for (int off = 16; off > 0; off >>= 1) d += __shfl_xor(d, off, 32);

    if (lane == 0) {
        const float num = d / (nrm[i] * nrm[i + 1]);         // no eps (faithful)
        pairterm[pair] = __logf(den[i]) + __logf(den[i + 1])
                       - 2.0f * num / TAU_F;
    }
}

// --- final scalar: fixed-order reduction of 4096 pair terms -----------------
__global__ __launch_bounds__(256) void sim_final_kernel(
    const float* __restrict__ pairterm, float* __restrict__ out) {
    float s = 0.0f;
    for (int p = threadIdx.x; p < N_ROWS / 2; p += 256) s += pairterm[p];
    __shared__ float red[256];
    red[threadIdx.x] = s;
    __syncthreads();
    #pragma unroll
    for (int off = 128; off > 0; off >>= 1) {
        if (threadIdx.x < off) red[threadIdx.x] += red[threadIdx.x + off];
        __syncthreads();
    }
    if (threadIdx.x == 0) out[0] = red[0] * (1.0f / (float)N_ROWS);
}

extern "C" void kernel_launch(void* const* d_in, const int* in_sizes, int n_in,
                              void* d_out, int out_size, void* d_ws, size_t ws_size,
                              hipStream_t stream) {
    (void)in_sizes; (void)n_in; (void)out_size; (void)ws_size;
    const float* batch = (const float*)d_in[0];
    float* out = (float*)d_out;

    float* x       = (float*)d_ws;                       // 8192*256 f32 = 8 MB
    float* nrm     = x + (size_t)N_ROWS * DIM;           // 8192 f32
    float* den     = nrm + N_ROWS;                       // 8192 f32
    float* partial = den + N_ROWS;                       // 8192*64 f32 = 2 MB
    float* pairs   = partial + (size_t)N_ROWS * NBLK;    // 4096 f32

    sim_reorder_norm_kernel<<<N_ROWS, DIM, 0, stream>>>(batch, x, nrm);

    dim3 grid(NBLK, NBLK);   // 64 x 64; lower-triangle blocks exit immediately
    sim_partial_kernel<<<grid, 256, 0, stream>>>(x, nrm, partial);

    sim_den_reduce_kernel<<<N_ROWS / 256, 256, 0, stream>>>(partial, den);
    sim_pair_kernel<<<N_ROWS / 16, 256, 0, stream>>>(x, nrm, den, pairs);
    sim_final_kernel<<<1, 256, 0, stream>>>(pairs, out);
}